// KMPNNGNN_43293270344036
// MI455X (gfx1250) — compile-verified
//
#include <hip/hip_runtime.h>
#include <math.h>

#define N_NODESC 15000
#define N_EDGESC 60000
#define EDGE_VOCABC 54
#define NODE_INDIMC 64
#define EDGE_INDIMC 32
#define HC 32
#define EHC 64
#define NUM_STEPSC 3
#define NEG_SLOPE 0.01f

typedef float v2f __attribute__((ext_vector_type(2)));
typedef float v8f __attribute__((ext_vector_type(8)));

static __device__ __forceinline__ v8f wmma_f32_16x16x4(v2f a, v2f b, v8f c) {
#if defined(__gfx1250__) && __has_builtin(__builtin_amdgcn_wmma_f32_16x16x4_f32)
  return __builtin_amdgcn_wmma_f32_16x16x4_f32(false, a, false, b, (short)0, c, false, false);
#else
  (void)a; (void)b;
  return c; // host pass / missing-builtin placeholder; device gfx1250 uses WMMA
#endif
}

static __device__ __forceinline__ float sigmoidf_(float x) { return 1.f / (1.f + expf(-x)); }

static __device__ __forceinline__ void atomicMaxFloat(float* addr, float val) {
  // Works given init to -inf: positive floats ordered as ints, negatives reverse-ordered as uints.
  if (val >= 0.f) atomicMax((int*)addr, __float_as_int(val));
  else            atomicMin((unsigned int*)addr, __float_as_uint(val));
}

// ---------------- counting sort of edges by edge_id (once) ----------------
__global__ __launch_bounds__(64) void k_zero_counts(int* counts) {
  counts[threadIdx.x] = 0;
}
__global__ __launch_bounds__(256) void k_count(const int* edge_ids, int* counts) {
  int e = blockIdx.x * 256 + threadIdx.x;
  if (e < N_EDGESC) atomicAdd(&counts[edge_ids[e]], 1);
}
__global__ void k_scan(const int* counts, int* offs) {
  if (threadIdx.x == 0) {
    int acc = 0;
    for (int t = 0; t < EDGE_VOCABC; ++t) { offs[t] = acc; acc += counts[t]; }
  }
}
__global__ __launch_bounds__(256) void k_scatter(const int* edge_ids, int* offs, int* perm) {
  int e = blockIdx.x * 256 + threadIdx.x;
  if (e < N_EDGESC) {
    int pos = atomicAdd(&offs[edge_ids[e]], 1);
    perm[pos] = e;
  }
}

// ---------------- per-type edge weight matrices: W_type[54][32][32] ----------------
__global__ __launch_bounds__(256) void k_type_weights(const float* edge_table,
    const float* e1W1, const float* e1b1, const float* e1W2, const float* e1b2,
    const float* e2W1, const float* e2b1, const float* e2W2, const float* e2b2,
    float* W_type) {
  int t = blockIdx.x;
  bool bond = t < 4;
  const float* W1 = bond ? e1W1 : e2W1;
  const float* b1 = bond ? e1b1 : e2b1;
  const float* W2 = bond ? e1W2 : e2W2;
  const float* b2 = bond ? e1b2 : e2b2;
  __shared__ float ef[EDGE_INDIMC];
  __shared__ float z[EHC];
  if (threadIdx.x < EDGE_INDIMC) ef[threadIdx.x] = edge_table[t * EDGE_INDIMC + threadIdx.x];
  __syncthreads();
  if (threadIdx.x < EHC) {
    float acc = b1[threadIdx.x];
    for (int k = 0; k < EDGE_INDIMC; ++k) acc += W1[threadIdx.x * EDGE_INDIMC + k] * ef[k];
    z[threadIdx.x] = acc > 0.f ? acc : 0.f;
  }
  __syncthreads();
  for (int o = threadIdx.x; o < HC * HC; o += 256) {
    float acc = b2[o];
    for (int k = 0; k < EHC; ++k) acc += W2[o * EHC + k] * z[k];
    W_type[t * (HC * HC) + o] = acc;
  }
}

// ---------------- node projection: h = relu(node_table[node_ids] @ proj_W^T + b), WMMA f32 ----------------
__global__ __launch_bounds__(256) void k_project(const int* node_ids, const float* node_table,
    const float* proj_W, const float* proj_b, float* h) {
  int wave = blockIdx.x * 8 + (threadIdx.x >> 5);
  int base = wave * 16;
  if (base >= N_NODESC) return;           // wave-uniform
  int lane = threadIdx.x & 31;
  int l = lane & 15;
  int halfk = (lane >> 4) * 2;            // K sub-offset 0 or 2
  int row = base + l;
  int nid = node_ids[row < N_NODESC ? row : (N_NODESC - 1)];
  const float* arow = node_table + nid * NODE_INDIMC;
  v8f c0 = {}; v8f c1 = {};
#pragma unroll
  for (int kk = 0; kk < NODE_INDIMC / 4; ++kk) {
    int k = kk * 4 + halfk;
    v2f a;  a[0]  = arow[k];                          a[1]  = arow[k + 1];
    v2f b0; b0[0] = proj_W[l * NODE_INDIMC + k];      b0[1] = proj_W[l * NODE_INDIMC + k + 1];
    v2f b1; b1[0] = proj_W[(16 + l) * NODE_INDIMC + k]; b1[1] = proj_W[(16 + l) * NODE_INDIMC + k + 1];
    c0 = wmma_f32_16x16x4(a, b0, c0);
    c1 = wmma_f32_16x16x4(a, b1, c1);
  }
  int mbase = (lane >> 4) * 8;            // C/D layout: lanes 0-15 -> rows i, 16-31 -> rows i+8
#pragma unroll
  for (int i = 0; i < 8; ++i) {
    int nrow = base + mbase + i;
    if (nrow < N_NODESC) {
      float v0 = c0[i] + proj_b[l];
      float v1 = c1[i] + proj_b[16 + l];
      h[nrow * HC + l]      = v0 > 0.f ? v0 : 0.f;
      h[nrow * HC + 16 + l] = v1 > 0.f ? v1 : 0.f;
    }
  }
}

// ---------------- per-step: node attention scalars + buffer init ----------------
__global__ __launch_bounds__(256) void k_node_prep(const float* h, const float* attn_w,
    float* sA, float* sB, float* maxb, float* sumb, float* agg) {
  int n = blockIdx.x * 256 + threadIdx.x;
  if (n >= N_NODESC) return;
  float a = 0.f, b = 0.f;
  for (int k = 0; k < HC; ++k) {
    float hv = h[n * HC + k];
    a += hv * attn_w[k];
    b += hv * attn_w[HC + k];
  }
  sA[n] = a; sB[n] = b;
  maxb[n] = -INFINITY; sumb[n] = 0.f;
  for (int k = 0; k < HC; ++k) agg[n * HC + k] = 0.f;
}

// ---------------- per-step: edge logits + segment max ----------------
__global__ __launch_bounds__(256) void k_edge_logits(const int* src, const int* dst,
    const float* sA, const float* sB, float* logits, float* maxb) {
  int e = blockIdx.x * 256 + threadIdx.x;
  if (e >= N_EDGESC) return;
  float a = sA[src[e]] + sB[dst[e]];
  a = a > 0.f ? a : NEG_SLOPE * a;        // leaky_relu(0.01)
  logits[e] = a;
  atomicMaxFloat(&maxb[dst[e]], a);
}

// ---------------- per-step: exp + segment sum ----------------
__global__ __launch_bounds__(256) void k_edge_exp(const int* dst, const float* logits,
    const float* maxb, float* ex, float* sumb) {
  int e = blockIdx.x * 256 + threadIdx.x;
  if (e >= N_EDGESC) return;
  float v = expf(logits[e] - maxb[dst[e]]);
  ex[e] = v;
  atomicAdd(&sumb[dst[e]], v);
}

// ---------------- per-step: type-grouped edge messages, WMMA f32, scatter-add ----------------
__global__ __launch_bounds__(256) void k_messages(const int* edge_ids, const int* src, const int* dst,
    const int* perm, const float* W_type, const float* h,
    const float* ex, const float* sumb, float* agg) {
  int wave = blockIdx.x * 8 + (threadIdx.x >> 5);
  int base = wave * 16;
  if (base >= N_EDGESC) return;           // wave-uniform
  int lane = threadIdx.x & 31;
  int l = lane & 15;
  int halfk = (lane >> 4) * 2;
  int lastIdx = (base + 15 < N_EDGESC) ? base + 15 : N_EDGESC - 1;
  int tFirst = edge_ids[perm[base]];
  int tLast  = edge_ids[perm[lastIdx]];
  if (tFirst == tLast && base + 15 < N_EDGESC) {
    // 16 edges, one shared 32x32 weight matrix -> 16x32 @ 32x32 GEMM via WMMA
    const float* Wt = W_type + tFirst * (HC * HC);
    int e = perm[base + l];
    const float* hs = h + src[e] * HC;
    v8f c0 = {}; v8f c1 = {};
#pragma unroll
    for (int kk = 0; kk < HC / 4; ++kk) {
      int k = kk * 4 + halfk;
      v2f a;  a[0]  = hs[k];                a[1]  = hs[k + 1];
      v2f b0; b0[0] = Wt[k * HC + l];       b0[1] = Wt[(k + 1) * HC + l];
      v2f b1; b1[0] = Wt[k * HC + 16 + l];  b1[1] = Wt[(k + 1) * HC + 16 + l];
      c0 = wmma_f32_16x16x4(a, b0, c0);
      c1 = wmma_f32_16x16x4(a, b1, c1);
    }
    int mbase = (lane >> 4) * 8;
#pragma unroll
    for (int i = 0; i < 8; ++i) {
      int em = perm[base + mbase + i];
      int d = dst[em];
      float al = ex[em] / sumb[d];
      atomicAdd(&agg[d * HC + l],      c0[i] * al);
      atomicAdd(&agg[d * HC + 16 + l], c1[i] * al);
    }
  } else {
    // mixed-type boundary group: scalar fallback (<=53 waves total)
    for (int m = 0; m < 16; ++m) {
      int idx = base + m;
      if (idx >= N_EDGESC) break;
      int em = perm[idx];
      const float* Wt = W_type + edge_ids[em] * (HC * HC);
      const float* hs = h + src[em] * HC;
      int d = dst[em];
      float al = ex[em] / sumb[d];
      float acc = 0.f;
      for (int k = 0; k < HC; ++k) acc += hs[k] * Wt[k * HC + lane];
      atomicAdd(&agg[d * HC + lane], acc * al);
    }
  }
}

// ---------------- per-step: GRU cell (one lane per (node, feature)) ----------------
__global__ __launch_bounds__(256) void k_gru(const float* agg, const float* hidden,
    const float* Wih, const float* Whh, const float* bih, const float* bhh, float* hout) {
  int tid = blockIdx.x * 256 + threadIdx.x;
  if (tid >= N_NODESC * HC) return;
  int n = tid >> 5;
  int j = tid & 31;
  float ir = bih[j], iz = bih[HC + j], in_ = bih[2 * HC + j];
  float hr = bhh[j], hz = bhh[HC + j], hn = bhh[2 * HC + j];
  for (int k = 0; k < HC; ++k) {
    float mv = agg[n * HC + k]; mv = mv > 0.f ? mv : 0.f;   // relu(agg)
    float hv = hidden[n * HC + k];
    ir += Wih[j * HC + k] * mv;
    iz += Wih[(HC + j) * HC + k] * mv;
    in_ += Wih[(2 * HC + j) * HC + k] * mv;
    hr += Whh[j * HC + k] * hv;
    hz += Whh[(HC + j) * HC + k] * hv;
    hn += Whh[(2 * HC + j) * HC + k] * hv;
  }
  float r = sigmoidf_(ir + hr);
  float z = sigmoidf_(iz + hz);
  float nn = tanhf(in_ + r * hn);
  hout[tid] = (1.f - z) * nn + z * hidden[tid];
}

extern "C" void kernel_launch(void* const* d_in, const int* in_sizes, int n_in,
                              void* d_out, int out_size, void* d_ws, size_t ws_size,
                              hipStream_t stream) {
  const int*   node_ids   = (const int*)d_in[0];
  const int*   edge_ids   = (const int*)d_in[1];
  const int*   src        = (const int*)d_in[2];
  const int*   dst        = (const int*)d_in[3];
  const float* node_table = (const float*)d_in[4];
  const float* edge_table = (const float*)d_in[5];
  const float* proj_W     = (const float*)d_in[6];
  const float* proj_b     = (const float*)d_in[7];
  const float* attn_w     = (const float*)d_in[8];
  const float* e1W1 = (const float*)d_in[9];
  const float* e1b1 = (const float*)d_in[10];
  const float* e1W2 = (const float*)d_in[11];
  const float* e1b2 = (const float*)d_in[12];
  const float* e2W1 = (const float*)d_in[13];
  const float* e2b1 = (const float*)d_in[14];
  const float* e2W2 = (const float*)d_in[15];
  const float* e2b2 = (const float*)d_in[16];
  const float* gWih = (const float*)d_in[17];
  const float* gWhh = (const float*)d_in[18];
  const float* gbih = (const float*)d_in[19];
  const float* gbhh = (const float*)d_in[20];

  float* ws = (float*)d_ws;
  float* h0     = ws; ws += N_NODESC * HC;
  float* h1     = ws; ws += N_NODESC * HC;
  float* Wt     = ws; ws += EDGE_VOCABC * HC * HC;
  float* sA     = ws; ws += N_NODESC;
  float* sB     = ws; ws += N_NODESC;
  float* maxb   = ws; ws += N_NODESC;
  float* sumb   = ws; ws += N_NODESC;
  float* agg    = ws; ws += N_NODESC * HC;
  float* logits = ws; ws += N_EDGESC;
  float* exv    = ws; ws += N_EDGESC;
  int*   perm   = (int*)ws; ws += N_EDGESC;
  int*   counts = (int*)ws; ws += 64;
  int*   offs   = (int*)ws; ws += 64;

  dim3 blk(256);
  int edgeBlocks = (N_EDGESC + 255) / 256;
  int nodeBlocks = (N_NODESC + 255) / 256;
  int projBlocks = (N_NODESC + 127) / 128;                  // 8 waves x 16 rows per block
  int msgBlocks  = ((N_EDGESC + 15) / 16 + 7) / 8;          // 8 waves x 16 edges per block
  int gruBlocks  = (N_NODESC * HC + 255) / 256;

  // one-time (per call) preprocessing
  k_zero_counts<<<1, 64, 0, stream>>>(counts);
  k_count<<<edgeBlocks, blk, 0, stream>>>(edge_ids, counts);
  k_scan<<<1, 1, 0, stream>>>(counts, offs);
  k_scatter<<<edgeBlocks, blk, 0, stream>>>(edge_ids, offs, perm);
  k_type_weights<<<EDGE_VOCABC, blk, 0, stream>>>(edge_table,
      e1W1, e1b1, e1W2, e1b2, e2W1, e2b1, e2W2, e2b2, Wt);
  k_project<<<projBlocks, blk, 0, stream>>>(node_ids, node_table, proj_W, proj_b, h0);

  float* hcur = h0;
  float* hnext = h1;
  for (int step = 0; step < NUM_STEPSC; ++step) {
    float* wout = (step == NUM_STEPSC - 1) ? (float*)d_out : hnext;
    k_node_prep<<<nodeBlocks, blk, 0, stream>>>(hcur, attn_w, sA, sB, maxb, sumb, agg);
    k_edge_logits<<<edgeBlocks, blk, 0, stream>>>(src, dst, sA, sB, logits, maxb);
    k_edge_exp<<<edgeBlocks, blk, 0, stream>>>(dst, logits, maxb, exv, sumb);
    k_messages<<<msgBlocks, blk, 0, stream>>>(edge_ids, src, dst, perm, Wt, hcur, exv, sumb, agg);
    k_gru<<<gruBlocks, blk, 0, stream>>>(agg, hcur, gWih, gWhh, gbih, gbhh, wout);
    float* tmp = hcur; hcur = hnext; hnext = tmp;
  }
}